// BatchAllTripletLoss_81707457839566
// MI455X (gfx1250) — compile-verified
//
#include <hip/hip_runtime.h>
#include <hip/hip_bf16.h>

// ---------------------------------------------------------------------------
// Batch-all triplet loss, n = 640, d = 128, MARGIN = 1.9, SQUARED = false.
// Stage 1: Gram via V_WMMA_F32_16X16X4_F32 (full fp32 on the matrix pipe).
// Stage 2: O(n^3) triplet reduction, LDS-cached, deterministic integer counts.
// ---------------------------------------------------------------------------

#define TL_N 640
#define TL_D 128
#define TL_TILES_PER_SIDE 40            // 640 / 16
#define TL_MARGIN 1.9f

typedef __attribute__((ext_vector_type(2))) float v2f;
typedef __attribute__((ext_vector_type(8))) float v8f;

// ---------------------------------------------------------------- norms ----
__global__ __launch_bounds__(256)
void tl_norms_kernel(const float* __restrict__ E, float* __restrict__ sq) {
    int i = blockIdx.x * blockDim.x + threadIdx.x;
    if (i < TL_N) {
        const float4* row = (const float4*)(E + (size_t)i * TL_D);
        float s = 0.0f;
#pragma unroll
        for (int k = 0; k < TL_D / 4; ++k) {
            float4 v = row[k];
            s += v.x * v.x + v.y * v.y + v.z * v.z + v.w * v.w;
        }
        sq[i] = s;
    }
}

// -------------------------------------------------- pairwise dist (WMMA) ----
// One wave per 16x16 output tile. 1600 tiles total; 8 waves/block, 200 blocks.
// A-matrix 16x4 f32 layout (per ISA): lanes 0-15 -> M=lane, VGPR0=K0,VGPR1=K1;
// lanes 16-31 -> M=lane-16, VGPR0=K2,VGPR1=K3. B (4x16) mirrors with N in lanes.
// For the Gram matrix (B[k][n] = E[col+n][k]) both operands use the identical
// per-lane load pattern.
__global__ __launch_bounds__(256)
void tl_dist_wmma_kernel(const float* __restrict__ E,
                         const float* __restrict__ sq,
                         float* __restrict__ dist) {
    const int wave = threadIdx.x >> 5;                 // 0..7
    const int lane = threadIdx.x & 31;
    const int tile = blockIdx.x * 8 + wave;            // 0..1599 (exact)
    const int tr = tile / TL_TILES_PER_SIDE;
    const int tc = tile - tr * TL_TILES_PER_SIDE;
    const int rowBase = tr * 16;
    const int colBase = tc * 16;

    const int m  = lane & 15;                          // matrix row/col in lane
    const int hi = lane >> 4;                          // 0: K0/K1, 1: K2/K3

    const float* arow = E + (size_t)(rowBase + m) * TL_D + hi * 2;
    const float* brow = E + (size_t)(colBase + m) * TL_D + hi * 2;

    v8f acc = {};
#pragma unroll
    for (int kk = 0; kk < TL_D / 4; ++kk) {
        v2f a = *(const v2f*)(arow + kk * 4);          // 8B-aligned
        v2f b = *(const v2f*)(brow + kk * 4);
        // D = A(16x4) * B(4x16) + C ; fp32 throughout.
        acc = __builtin_amdgcn_wmma_f32_16x16x4_f32(
            /*neg_a=*/false, a, /*neg_b=*/false, b,
            /*c_mod=*/(short)0, acc, /*reuse_a=*/false, /*reuse_b=*/false);
    }

    // C/D layout: VGPR r -> M = r + 8*hi (per half-wave), N = lane%16.
    const float sqj = sq[colBase + m];
#pragma unroll
    for (int r = 0; r < 8; ++r) {
        const int i = rowBase + r + hi * 8;
        float d = sq[i] - 2.0f * acc[r] + sqj;
        d = d > 0.0f ? d : 0.0f;                       // relu
        // sqrt(d + (d==0)*1e-16) * (d!=0)  ==  d==0 ? 0 : sqrt(d)
        float dd = (d == 0.0f) ? 0.0f : __builtin_sqrtf(d);
        dist[(size_t)i * TL_N + colBase + m] = dd;
    }
}

// ----------------------------------------------------------- triplet pass ----
// One block per anchor i. dist row + labels in LDS. Exact integer counts
// (deterministic, no float atomics).
__global__ __launch_bounds__(256)
void tl_triplet_kernel(const float* __restrict__ dist,
                       const int* __restrict__ lab,
                       float* __restrict__ partTot,
                       unsigned* __restrict__ partNE,
                       unsigned* __restrict__ partV) {
    __shared__ float    sd[TL_N];
    __shared__ int      sl[TL_N];
    __shared__ float    rT[256];
    __shared__ unsigned rC[256];

    const int i = blockIdx.x;
    const int t = threadIdx.x;

    for (int j = t; j < TL_N; j += 256) {
        sd[j] = dist[(size_t)i * TL_N + j];
        sl[j] = lab[j];
    }
    __syncthreads();

    const int li = sl[i];

    float    tot = 0.0f;
    unsigned cnt = 0;
    for (int j = t; j < TL_N; j += 256) {
        if (j == i || sl[j] != li) continue;           // j must be positive
        const float dij = sd[j];
        for (int k = 0; k < TL_N; ++k) {
            if (sl[k] == li) continue;                 // k must be negative
            const float tr = dij - sd[k] + TL_MARGIN;
            if (tr > 0.0f) { tot += tr; ++cnt; }
        }
    }

    // positive / negative counts for num_valid = npos * nneg
    unsigned np = 0, nn = 0;
    for (int j = t; j < TL_N; j += 256) {
        np += (unsigned)(j != i && sl[j] == li);
        nn += (unsigned)(sl[j] != li);
    }

    // reduce (tot, cnt)
    rT[t] = tot; rC[t] = cnt;
    __syncthreads();
    for (int s = 128; s > 0; s >>= 1) {
        if (t < s) { rT[t] += rT[t + s]; rC[t] += rC[t + s]; }
        __syncthreads();
    }
    const float    blkTot = rT[0];
    const unsigned blkCnt = rC[0];
    __syncthreads();

    // reduce packed (np<<16 | nn): totals < 2^16 each, no cross-field carry.
    rC[t] = (np << 16) | nn;
    __syncthreads();
    for (int s = 128; s > 0; s >>= 1) {
        if (t < s) rC[t] += rC[t + s];
        __syncthreads();
    }

    if (t == 0) {
        const unsigned packed = rC[0];
        const unsigned nposT = packed >> 16;
        const unsigned nnegT = packed & 0xFFFFu;
        partTot[i] = blkTot;
        partNE[i]  = blkCnt;
        partV[i]   = nposT * nnegT;
    }
}

// -------------------------------------------------------------- finalize ----
__global__ __launch_bounds__(256)
void tl_finalize_kernel(const float* __restrict__ partTot,
                        const unsigned* __restrict__ partNE,
                        const unsigned* __restrict__ partV,
                        float* __restrict__ out) {
    __shared__ float              sT[256];
    __shared__ unsigned long long sN[256];
    __shared__ unsigned long long sV[256];

    const int t = threadIdx.x;
    float tot = 0.0f;
    unsigned long long ne = 0ull, va = 0ull;
    for (int i = t; i < TL_N; i += 256) {
        tot += partTot[i];
        ne  += partNE[i];
        va  += partV[i];
    }
    sT[t] = tot; sN[t] = ne; sV[t] = va;
    __syncthreads();
    for (int s = 128; s > 0; s >>= 1) {
        if (t < s) { sT[t] += sT[t + s]; sN[t] += sN[t + s]; sV[t] += sV[t + s]; }
        __syncthreads();
    }
    if (t == 0) {
        const float total = sT[0];
        const float nne   = (float)sN[0];
        const float nv    = (float)sV[0];
        const float loss  = (nne > 0.0f) ? total / fmaxf(nne, 1.0f) : 0.0f;
        out[0] = loss;                       // loss
        out[1] = nv;                         // num_valid
        out[2] = nne;                        // num_non_easy
        out[3] = nne / (nv + 1e-16f);        // frac
    }
}

// ---------------------------------------------------------------------------
extern "C" void kernel_launch(void* const* d_in, const int* in_sizes, int n_in,
                              void* d_out, int out_size, void* d_ws, size_t ws_size,
                              hipStream_t stream) {
    const float* E   = (const float*)d_in[0];   // [8,80,128] f32 -> [640,128]
    const int*   lab = (const int*)d_in[1];     // [640] int
    float*       out = (float*)d_out;           // 4 scalars

    char* ws = (char*)d_ws;
    float*    sq      = (float*)ws;                                   // 2560 B
    float*    dist    = (float*)(ws + 4096);                          // 1.6 MB
    float*    partTot = (float*)(ws + 4096 + (size_t)TL_N * TL_N * 4);
    unsigned* partNE  = (unsigned*)((char*)partTot + TL_N * 4);
    unsigned* partV   = (unsigned*)((char*)partNE  + TL_N * 4);

    tl_norms_kernel<<<(TL_N + 255) / 256, 256, 0, stream>>>(E, sq);
    tl_dist_wmma_kernel<<<200, 256, 0, stream>>>(E, sq, dist);   // 200*8 = 1600 tiles
    tl_triplet_kernel<<<TL_N, 256, 0, stream>>>(dist, lab, partTot, partNE, partV);
    tl_finalize_kernel<<<1, 256, 0, stream>>>(partTot, partNE, partV, out);
}